// TemporalBiasMHA_43576738185820
// MI455X (gfx1250) — compile-verified
//
#include <hip/hip_runtime.h>
#include <math.h>

// Problem constants (match reference)
#define BB 4
#define WW 2048
#define DD 512
#define HH 8
#define DKK 64          // DD / HH
#define BIAS_SCALE 4.0f
#define LN_EPS 1e-5f
#define SOFT_SCALE 0.125f   // 1/sqrt(64)

typedef __attribute__((ext_vector_type(2))) float v2f;
typedef __attribute__((ext_vector_type(8))) float v8f;

// D = A(16x4) * B(4x16) + C(16x16), fp32 WMMA (CDNA5 V_WMMA_F32_16X16X4_F32)
__device__ __forceinline__ v8f wmma_f32(v2f a, v2f b, v8f c) {
    return __builtin_amdgcn_wmma_f32_16x16x4_f32(
        /*neg_a=*/false, a, /*neg_b=*/false, b,
        /*c_mod=*/(short)0, c, /*reuse_a=*/false, /*reuse_b=*/false);
}

// ---------------------------------------------------------------------------
// Kernel 1: fused QKV projection.  out[b][h][w][dk] = x[b][w]·W[:,n] + bias[n]
// Block: 128 threads (4 waves).  Block tile: 64(M) x 16(N), K staged in LDS.
// ---------------------------------------------------------------------------
#define KC 32
__global__ __launch_bounds__(128) void qkv_proj_kernel(
    const float* __restrict__ x,
    const float* __restrict__ Wq, const float* __restrict__ Wk, const float* __restrict__ Wv,
    const float* __restrict__ bq, const float* __restrict__ bk, const float* __restrict__ bv,
    float* __restrict__ q_ws, float* __restrict__ k_ws, float* __restrict__ v_ws)
{
    const int m0 = blockIdx.x * 64;
    const int n0 = blockIdx.y * 16;
    const int zb = blockIdx.z;              // b*3 + which
    const int b = zb / 3, which = zb - 3 * b;

    const float* Wm   = (which == 0) ? Wq : (which == 1) ? Wk : Wv;
    const float* bias = (which == 0) ? bq : (which == 1) ? bk : bv;
    float*       outp = (which == 0) ? q_ws : (which == 1) ? k_ws : v_ws;

    __shared__ float Alds[64][36];          // 64 x KC, +4 pad (keeps 16B align)
    __shared__ float Blds[KC][16];

    const int tid  = threadIdx.x;
    const int lane = tid & 31;
    const int wav  = tid >> 5;
    const int half = lane >> 4;
    const int l    = lane & 15;

    v8f acc = {};
    const float* xb = x + (size_t)b * WW * DD;

    for (int k0 = 0; k0 < DD; k0 += KC) {
        // cooperative load A tile: 64x32 floats as float4 (coalesced)
        #pragma unroll
        for (int i = 0; i < 4; ++i) {
            int f4  = i * 128 + tid;
            int row = f4 >> 3, c4 = f4 & 7;
            float4 v = *(const float4*)(xb + (size_t)(m0 + row) * DD + k0 + c4 * 4);
            *(float4*)(&Alds[row][c4 * 4]) = v;
        }
        // cooperative load B tile: 32x16 floats (one float4 per thread)
        {
            int row = tid >> 2, c4 = tid & 3;
            float4 v = *(const float4*)(Wm + (size_t)(k0 + row) * DD + n0 + c4 * 4);
            *(float4*)(&Blds[row][c4 * 4]) = v;
        }
        if (k0 + KC < DD)   // hint next A tile into cache (global_prefetch_b8)
            __builtin_prefetch(xb + (size_t)(m0 + (tid >> 1)) * DD + k0 + KC, 0, 1);
        __syncthreads();

        const int arow = wav * 16 + l;
        #pragma unroll
        for (int kk = 0; kk < KC; kk += 4) {
            v2f a, bf;
            a.x  = Alds[arow][kk + 2 * half];
            a.y  = Alds[arow][kk + 2 * half + 1];
            bf.x = Blds[kk + 2 * half][l];
            bf.y = Blds[kk + 2 * half + 1][l];
            acc  = wmma_f32(a, bf, acc);
        }
        __syncthreads();
    }

    // store to head-major layout (B, H, W, DK)
    const int col = n0 + l;
    const int h   = col >> 6;
    const int dk  = col & 63;
    const float bv_ = bias[col];
    #pragma unroll
    for (int i = 0; i < 8; ++i) {
        int row = m0 + wav * 16 + i + 8 * half;
        outp[(((size_t)b * HH + h) * WW + row) * DKK + dk] = acc[i] + bv_;
    }
}

// ---------------------------------------------------------------------------
// Kernel 2: flash attention with temporal bias + mask.
// Block: 128 threads (4 waves); each wave owns one 16-query tile; the 4 waves
// share (b,h), so K/V 16x64 tiles are double-buffered in LDS via CDNA5
// GLOBAL_LOAD_ASYNC_TO_LDS_B128 (ASYNCcnt) and consumed by all 4 waves.
// ---------------------------------------------------------------------------
#define KV_STRIDE 68   // 64 + 4 pad: 4-bank lane skew, rows stay 8B aligned

// async copy of one contiguous 16x64-float tile (4KB) into padded LDS rows
__device__ __forceinline__ void async_tile_to_lds(const float* __restrict__ src,
                                                  unsigned lds_base_bytes, int tid) {
    #pragma unroll
    for (int i = 0; i < 2; ++i) {
        int f   = (i * 128 + tid) * 4;          // first float of this 16B chunk
        int row = f >> 6, col = f & 63;
        unsigned lds_off = lds_base_bytes + (unsigned)(row * KV_STRIDE + col) * 4u;
        const float* g = src + f;
        asm volatile("global_load_async_to_lds_b128 %0, %1, off"
                     :: "v"(lds_off), "v"(g) : "memory");
    }
}

__global__ __launch_bounds__(128) void flash_attn_kernel(
    const float* __restrict__ q_ws, const float* __restrict__ k_ws,
    const float* __restrict__ v_ws,
    const float* __restrict__ E_time, const int* __restrict__ A_time,
    float* __restrict__ y_ws)
{
    const int tid  = threadIdx.x;
    const int wav  = tid >> 5;
    const int lane = tid & 31;
    const int half = lane >> 4;
    const int l    = lane & 15;

    const int qt = blockIdx.x * 4 + wav;     // 0..127
    const int h  = blockIdx.y;
    const int b  = blockIdx.z;
    const int q0 = qt * 16;

    __shared__ float Klds[2][16][KV_STRIDE];
    __shared__ float Vlds[2][16][KV_STRIDE];
    __shared__ float Plds[4][16][18];        // per-wave P staging (8B rows)

    const size_t head_base = ((size_t)b * HH + h) * WW;
    const float* kbase = k_ws + head_base * DKK;
    const float* vbase = v_ws + head_base * DKK;
    const unsigned klds0 = (unsigned)(size_t)&Klds[0][0][0];
    const unsigned vlds0 = (unsigned)(size_t)&Vlds[0][0][0];
    const unsigned kvbuf_bytes = 16u * KV_STRIDE * 4u;

    // preload Q fragments (A-operand layout) for all 16 K-steps of DK=64
    v2f qfrag[16];
    {
        const float* qrow = q_ws + (head_base + q0 + l) * DKK;
        #pragma unroll
        for (int kk4 = 0; kk4 < 16; ++kk4) {
            qfrag[kk4].x = qrow[kk4 * 4 + 2 * half];
            qfrag[kk4].y = qrow[kk4 * 4 + 2 * half + 1];
        }
    }

    v8f o0 = {}, o1 = {}, o2 = {}, o3 = {};
    float mrun[8], Lrun[8];
    #pragma unroll
    for (int i = 0; i < 8; ++i) { mrun[i] = -1e37f; Lrun[i] = 0.0f; }

    // prime the pipeline: async-stage tile 0 into buffer 0
    async_tile_to_lds(kbase, klds0, tid);
    async_tile_to_lds(vbase, vlds0, tid);

    for (int j = 0; j < WW / 16; ++j) {
        const int buf = j & 1;

        // issue next tile's async copies into the other buffer (its previous
        // readers finished at the trailing barrier of iteration j-1)
        if (j + 1 < WW / 16) {
            const float* kn = kbase + (size_t)(j + 1) * 16 * DKK;
            const float* vn = vbase + (size_t)(j + 1) * 16 * DKK;
            unsigned nb = (unsigned)((j + 1) & 1) * kvbuf_bytes;
            async_tile_to_lds(kn, klds0 + nb, tid);
            async_tile_to_lds(vn, vlds0 + nb, tid);
            // async loads complete in order: first 4 (tile j) done when cnt<=4
            asm volatile("s_wait_asynccnt 0x4" ::: "memory");
        } else {
            asm volatile("s_wait_asynccnt 0x0" ::: "memory");
        }
        __syncthreads();                     // tile j visible to all waves

        // ----- S = Q * K^T (16x16), K-dim = 64 -> 16 WMMAs -----
        v8f s = {};
        #pragma unroll
        for (int kk4 = 0; kk4 < 16; ++kk4) {
            v2f bf;
            bf.x = Klds[buf][l][kk4 * 4 + 2 * half];
            bf.y = Klds[buf][l][kk4 * 4 + 2 * half + 1];
            s = wmma_f32(qfrag[kk4], bf, s);
        }

        // ----- bias + mask (each E/A element read exactly once, coalesced) --
        const int kcol = j * 16 + l;
        #pragma unroll
        for (int i = 0; i < 8; ++i) {
            int row = q0 + i + 8 * half;
            size_t idx = ((size_t)b * WW + row) * WW + kcol;
            float e = E_time[idx];
            int   a = A_time[idx];
            float biasv = fminf(fmaxf(e, -1.0f), 1.0f) * BIAS_SCALE;
            float sv = s[i] * SOFT_SCALE + biasv;
            s[i] = (a > 0) ? sv : -1e38f;
        }

        // ----- online softmax (row stats replicated across 16-lane halves) --
        float p[8], sc[8];
        #pragma unroll
        for (int i = 0; i < 8; ++i) {
            float rm = s[i];
            rm = fmaxf(rm, __shfl_xor(rm, 1, 32));
            rm = fmaxf(rm, __shfl_xor(rm, 2, 32));
            rm = fmaxf(rm, __shfl_xor(rm, 4, 32));
            rm = fmaxf(rm, __shfl_xor(rm, 8, 32));
            float mn = fmaxf(mrun[i], rm);
            sc[i] = __expf(mrun[i] - mn);
            float pv = __expf(s[i] - mn);
            p[i] = pv;
            float rs = pv;
            rs += __shfl_xor(rs, 1, 32);
            rs += __shfl_xor(rs, 2, 32);
            rs += __shfl_xor(rs, 4, 32);
            rs += __shfl_xor(rs, 8, 32);
            Lrun[i] = Lrun[i] * sc[i] + rs;
            mrun[i] = mn;
        }
        #pragma unroll
        for (int i = 0; i < 8; ++i) {
            o0[i] *= sc[i]; o1[i] *= sc[i]; o2[i] *= sc[i]; o3[i] *= sc[i];
        }

        // ----- C-layout -> A-layout via LDS, then O += P * V -----
        #pragma unroll
        for (int i = 0; i < 8; ++i)
            Plds[wav][i + 8 * half][l] = p[i];
        // same-wave DS ops are in order; loads below observe the stores

        #pragma unroll
        for (int kk = 0; kk < 16; kk += 4) {
            v2f a;
            a.x = Plds[wav][l][kk + 2 * half];
            a.y = Plds[wav][l][kk + 2 * half + 1];
            const int r0 = kk + 2 * half;
            v2f bf;
            bf.x = Vlds[buf][r0][l];      bf.y = Vlds[buf][r0 + 1][l];
            o0 = wmma_f32(a, bf, o0);
            bf.x = Vlds[buf][r0][16 + l]; bf.y = Vlds[buf][r0 + 1][16 + l];
            o1 = wmma_f32(a, bf, o1);
            bf.x = Vlds[buf][r0][32 + l]; bf.y = Vlds[buf][r0 + 1][32 + l];
            o2 = wmma_f32(a, bf, o2);
            bf.x = Vlds[buf][r0][48 + l]; bf.y = Vlds[buf][r0 + 1][48 + l];
            o3 = wmma_f32(a, bf, o3);
        }
        __syncthreads();   // all waves done reading buf before it is restaged
    }

    // ----- normalize and store y (merge heads -> (B, W, D) layout) -----
    #pragma unroll
    for (int i = 0; i < 8; ++i) {
        float inv = 1.0f / Lrun[i];
        int row = q0 + i + 8 * half;
        float* yr = y_ws + ((size_t)b * WW + row) * DD + h * DKK;
        yr[l]      = o0[i] * inv;
        yr[16 + l] = o1[i] * inv;
        yr[32 + l] = o2[i] * inv;
        yr[48 + l] = o3[i] * inv;
    }
}

// ---------------------------------------------------------------------------
// Kernel 3: output projection + residual + LayerNorm, fused.
// Block: 256 threads (8 waves) computes 16 full rows (16 x 512).
// ---------------------------------------------------------------------------
__global__ __launch_bounds__(256) void out_ln_kernel(
    const float* __restrict__ y_ws, const float* __restrict__ Wo,
    const float* __restrict__ bo,   const float* __restrict__ x,
    const float* __restrict__ ln_g, const float* __restrict__ ln_b,
    float* __restrict__ out)
{
    const int m0 = blockIdx.x * 16;
    const int b  = blockIdx.y;
    const int tid  = threadIdx.x;
    const int wav  = tid >> 5;
    const int lane = tid & 31;
    const int half = lane >> 4;
    const int l    = lane & 15;

    __shared__ float Alds[16][36];
    __shared__ float Hlds[16][520];          // 16 x 512 (+8 pad)

    v8f a0 = {}, a1 = {}, a2 = {}, a3 = {};
    const float* yb = y_ws + (size_t)b * WW * DD;

    for (int k0 = 0; k0 < DD; k0 += KC) {
        if (tid < 128) {                     // 16x32 A tile, one float4/thread
            int row = tid >> 3, c4 = tid & 7;
            *(float4*)(&Alds[row][c4 * 4]) =
                *(const float4*)(yb + (size_t)(m0 + row) * DD + k0 + c4 * 4);
        }
        __syncthreads();
        #pragma unroll
        for (int kk = 0; kk < KC; kk += 4) {
            v2f a;
            a.x = Alds[l][kk + 2 * half];
            a.y = Alds[l][kk + 2 * half + 1];
            const float* wr = Wo + (size_t)(k0 + kk + 2 * half) * DD + wav * 64 + l;
            v2f bf;
            bf.x = wr[0];   bf.y = wr[DD];        a0 = wmma_f32(a, bf, a0);
            bf.x = wr[16];  bf.y = wr[DD + 16];   a1 = wmma_f32(a, bf, a1);
            bf.x = wr[32];  bf.y = wr[DD + 32];   a2 = wmma_f32(a, bf, a2);
            bf.x = wr[48];  bf.y = wr[DD + 48];   a3 = wmma_f32(a, bf, a3);
        }
        __syncthreads();
    }

    // h = proj + bias + residual -> LDS
    const float* xb = x + ((size_t)b * WW + m0) * DD;
    #pragma unroll
    for (int i = 0; i < 8; ++i) {
        int row = i + 8 * half;
        int n = wav * 64 + l;
        Hlds[row][n]      = a0[i] + bo[n]      + xb[(size_t)row * DD + n];
        Hlds[row][n + 16] = a1[i] + bo[n + 16] + xb[(size_t)row * DD + n + 16];
        Hlds[row][n + 32] = a2[i] + bo[n + 32] + xb[(size_t)row * DD + n + 32];
        Hlds[row][n + 48] = a3[i] + bo[n + 48] + xb[(size_t)row * DD + n + 48];
    }
    __syncthreads();

    // LayerNorm: 16 threads per row, 32 cols each; shuffle-reduce within halves
    const int row = tid >> 4;
    const int seg = tid & 15;
    float s1 = 0.0f, s2 = 0.0f;
    #pragma unroll
    for (int c = 0; c < 32; ++c) {
        float v = Hlds[row][seg * 32 + c];
        s1 += v; s2 += v * v;
    }
    s1 += __shfl_xor(s1, 1, 32);  s2 += __shfl_xor(s2, 1, 32);
    s1 += __shfl_xor(s1, 2, 32);  s2 += __shfl_xor(s2, 2, 32);
    s1 += __shfl_xor(s1, 4, 32);  s2 += __shfl_xor(s2, 4, 32);
    s1 += __shfl_xor(s1, 8, 32);  s2 += __shfl_xor(s2, 8, 32);
    const float mu  = s1 * (1.0f / DD);
    const float var = s2 * (1.0f / DD) - mu * mu;
    const float r   = rsqrtf(var + LN_EPS);

    float* orow = out + ((size_t)b * WW + m0 + row) * DD;
    #pragma unroll
    for (int c = 0; c < 32; ++c) {
        int n = seg * 32 + c;
        orow[n] = (Hlds[row][n] - mu) * r * ln_g[n] + ln_b[n];
    }
}

// ---------------------------------------------------------------------------
extern "C" void kernel_launch(void* const* d_in, const int* in_sizes, int n_in,
                              void* d_out, int out_size, void* d_ws, size_t ws_size,
                              hipStream_t stream) {
    const float* x      = (const float*)d_in[0];
    const float* E_time = (const float*)d_in[1];
    const int*   A_time = (const int*)  d_in[2];
    const float* Wq     = (const float*)d_in[3];
    const float* Wk     = (const float*)d_in[4];
    const float* Wv     = (const float*)d_in[5];
    const float* Wo     = (const float*)d_in[6];
    const float* bq     = (const float*)d_in[7];
    const float* bk     = (const float*)d_in[8];
    const float* bv     = (const float*)d_in[9];
    const float* bo     = (const float*)d_in[10];
    const float* ln_g   = (const float*)d_in[11];
    const float* ln_b   = (const float*)d_in[12];
    float* out = (float*)d_out;

    const size_t HEADSZ = (size_t)BB * HH * WW * DKK;   // 4.19M floats each
    float* q_ws = (float*)d_ws;
    float* k_ws = q_ws + HEADSZ;
    float* v_ws = k_ws + HEADSZ;
    float* y_ws = v_ws + HEADSZ;

    // 1) QKV projections: grid (M/64, N/16, B*3)
    qkv_proj_kernel<<<dim3(WW / 64, DD / 16, BB * 3), 128, 0, stream>>>(
        x, Wq, Wk, Wv, bq, bk, bv, q_ws, k_ws, v_ws);

    // 2) Flash attention: grid (qtiles/4, H, B), 4 waves/block
    flash_attn_kernel<<<dim3(WW / 16 / 4, HH, BB), 128, 0, stream>>>(
        q_ws, k_ws, v_ws, E_time, A_time, y_ws);

    // 3) Output projection + residual + LayerNorm: grid (W/16, B)
    out_ln_kernel<<<dim3(WW / 16, BB), 256, 0, stream>>>(
        y_ws, Wo, bo, x, ln_g, ln_b, out);
}